// FC_Caps_54803782697366
// MI455X (gfx1250) — compile-verified
//
#include <hip/hip_runtime.h>

// ---------------------------------------------------------------------------
// FC-Caps dynamic routing, fused, fp32 WMMA (V_WMMA_F32_16X16X4_F32).
// B=32, I=1024, O=64, D_in=16, D_out=32, 3 routing iterations.
//
// Strategy: never materialize u_hat (256MB). Recompute it from W (128MB,
// L2-resident on MI455X's 192MB L2) once per routing iteration, fusing
// a = u_hat.v, logit update, softmax over O (local per (b,i)), and the
// weighted accumulation s = sum_i c*u_hat into a single pass.
// ---------------------------------------------------------------------------

typedef __attribute__((ext_vector_type(2))) float v2f;
typedef __attribute__((ext_vector_type(8))) float v8f;

#define B_TOT 32
#define I_TOT 1024
#define O_TOT 64
#define DIN   16
#define DOUT  32
#define OD    2048      // O_TOT * DOUT
#define IT    8         // capsules i per block

// LDS partition (floats)
#define SPART_F 32768   // s_partial [16 b][64 o][32 d]
#define VLDS_F  32768   // v_prev    [16 b][64 o][32 d]
#define ABUF_F  1024    // a_buf     [64 o][16 b]

// MODE 0: iter0 (uniform c = 1/64), no v/logits.
// MODE 1: iter1: a0 = u_hat.v0 ; logits = a0 ; c = softmax(a0) ; s1 += c*u_hat
// MODE 2: iter2: a  = logits + u_hat.v1 ; c = softmax(a) ; s2 += c*u_hat
template<int MODE>
__global__ __launch_bounds__(256, 1)
void caps_pass(const float* __restrict__ x, const float* __restrict__ W,
               const float* __restrict__ v_prev, float* __restrict__ logits,
               float* __restrict__ s_out)
{
    extern __shared__ float smem[];
    float* s_part = smem;                       // SPART_F
    float* v_lds  = smem + SPART_F;             // VLDS_F  (MODE 1/2)
    float* a_buf  = smem + SPART_F + VLDS_F;    // ABUF_F  (MODE 1/2)

    const int tid   = threadIdx.x;
    const int wv    = tid >> 5;
    const int lane  = tid & 31;
    const int bl    = lane & 15;        // local batch index (WMMA N / col)
    const int half  = lane >> 4;        // wave half (ISA lane-split)
    const int b0    = blockIdx.y * 16;  // batch group base
    const int i0    = blockIdx.x * IT;  // capsule-i tile base
    const int obase = wv * 8;           // this wave owns o = obase..obase+7

    for (int k = tid; k < SPART_F; k += 256) s_part[k] = 0.0f;
    if constexpr (MODE != 0) {
        for (int k = tid; k < VLDS_F; k += 256) v_lds[k] = v_prev[b0 * OD + k];
    }
    __syncthreads();

    for (int ii = 0; ii < IT; ++ii) {
        const int i = i0 + ii;

        // --- B fragments (x^T, 4 x K=4 chunks): lane -> col b = bl,
        //     VGPR0 row K = 2*half, VGPR1 row K = 2*half+1
        v2f bfrag[4];
        const float* xp = x + ((size_t)(b0 + bl) * I_TOT + i) * DIN;
        #pragma unroll
        for (int kk = 0; kk < 4; ++kk) {
            bfrag[kk] = *(const v2f*)(xp + kk * 4 + 2 * half);
        }

        // --- u_hat tiles via WMMA: D[16 od x 16 b], 2 d-halves per o
        v8f acc[8][2];
        #pragma unroll
        for (int oo = 0; oo < 8; ++oo) {
            const int o = obase + oo;
            #pragma unroll
            for (int t = 0; t < 2; ++t) {
                const int od_row = o * DOUT + t * 16 + bl;  // A row M = lane%16
                const float* wp = W + ((size_t)i * OD + od_row) * DIN;
                v8f u = {};
                #pragma unroll
                for (int kk = 0; kk < 4; ++kk) {
                    // A fragment: VGPR0 = K(2*half), VGPR1 = K(2*half+1)
                    v2f afrag = *(const v2f*)(wp + kk * 4 + 2 * half);
                    u = __builtin_amdgcn_wmma_f32_16x16x4_f32(
                        false, afrag, false, bfrag[kk], (short)0, u, false, false);
                }
                acc[oo][t] = u;
            }
        }
        // D-tile element (VGPR r, lane): u_hat[o, d = t*16 + r + 8*half, b = bl]

        if constexpr (MODE == 0) {
            constexpr float cu = 1.0f / 64.0f;
            #pragma unroll
            for (int oo = 0; oo < 8; ++oo) {
                const int o = obase + oo;
                #pragma unroll
                for (int t = 0; t < 2; ++t)
                    #pragma unroll
                    for (int r = 0; r < 8; ++r) {
                        const int d = t * 16 + r + half * 8;
                        s_part[bl * OD + o * DOUT + d] += cu * acc[oo][t][r];
                    }
            }
        } else {
            // agreement a[b,o] = sum_d u_hat * v_prev
            float aval[8];
            #pragma unroll
            for (int oo = 0; oo < 8; ++oo) {
                const int o = obase + oo;
                float s = 0.0f;
                #pragma unroll
                for (int t = 0; t < 2; ++t)
                    #pragma unroll
                    for (int r = 0; r < 8; ++r) {
                        const int d = t * 16 + r + half * 8;
                        s += acc[oo][t][r] * v_lds[bl * OD + o * DOUT + d];
                    }
                s += __shfl_xor(s, 16, 32);  // combine wave halves (d-split)
                const size_t lidx = ((size_t)(b0 + bl) * I_TOT + i) * O_TOT + o;
                if constexpr (MODE == 2) s += logits[lidx];
                if constexpr (MODE == 1) { if (lane < 16) logits[lidx] = s; }
                aval[oo] = s;
                if (lane < 16) a_buf[o * 16 + bl] = s;
            }
            __syncthreads();

            // softmax over the 64 output capsules (local to (b,i))
            float m = -1e30f;
            for (int o = 0; o < O_TOT; ++o) m = fmaxf(m, a_buf[o * 16 + bl]);
            float den = 0.0f;
            for (int o = 0; o < O_TOT; ++o) den += __expf(a_buf[o * 16 + bl] - m);
            const float inv = 1.0f / den;
            __syncthreads();  // a_buf reusable next i

            #pragma unroll
            for (int oo = 0; oo < 8; ++oo) {
                const int o = obase + oo;
                const float c = __expf(aval[oo] - m) * inv;
                #pragma unroll
                for (int t = 0; t < 2; ++t)
                    #pragma unroll
                    for (int r = 0; r < 8; ++r) {
                        const int d = t * 16 + r + half * 8;
                        s_part[bl * OD + o * DOUT + d] += c * acc[oo][t][r];
                    }
            }
        }
    }

    __syncthreads();
    for (int k = tid; k < SPART_F; k += 256)
        atomicAdd(&s_out[b0 * OD + k], s_part[k]);
}

__global__ void caps_squash(const float* __restrict__ s_in,
                            const float* __restrict__ bias,
                            float* __restrict__ v_out)
{
    const int p = blockIdx.x * blockDim.x + threadIdx.x;  // p = b*64 + o
    if (p >= B_TOT * O_TOT) return;
    const int o = p & 63;
    const float* sp = s_in + (size_t)p * DOUT;
    float buf[DOUT];
    float dot = 0.0f;
    #pragma unroll
    for (int d = 0; d < DOUT; ++d) {
        float t = sp[d] + (bias ? bias[o * DOUT + d] : 0.0f);
        buf[d] = t;
        dot += t * t;
    }
    const float scale = dot / (1.0f + dot) * rsqrtf(dot + 1e-8f);
    float* vp = v_out + (size_t)p * DOUT;
    #pragma unroll
    for (int d = 0; d < DOUT; ++d) vp[d] = scale * buf[d];
}

__global__ void caps_zero(float* __restrict__ p, int n)
{
    const int k = blockIdx.x * blockDim.x + threadIdx.x;
    if (k < n) p[k] = 0.0f;
}

extern "C" void kernel_launch(void* const* d_in, const int* in_sizes, int n_in,
                              void* d_out, int out_size, void* d_ws, size_t ws_size,
                              hipStream_t stream)
{
    const float* x    = (const float*)d_in[0];  // [32,1024,16]
    const float* W    = (const float*)d_in[1];  // [1,1024,64,32,16]
    const float* bias = (const float*)d_in[2];  // [1,1,64,32]
    float* out = (float*)d_out;                 // [32,64,32]

    float* ws = (float*)d_ws;
    float* s0     = ws;                 // 65536
    float* s1     = ws + 65536;         // 65536
    float* s2     = ws + 131072;        // 65536
    float* v0     = ws + 196608;        // 65536
    float* v1     = ws + 262144;        // 65536
    float* logits = ws + 327680;        // 2,097,152  (b_ij accumulated)

    // zero s0,s1,s2 (contiguous)
    caps_zero<<<(3 * 65536 + 255) / 256, 256, 0, stream>>>(s0, 3 * 65536);

    dim3 grid(I_TOT / IT, B_TOT / 16);
    const size_t smem0  = (size_t)SPART_F * sizeof(float);
    const size_t smem12 = (size_t)(SPART_F + VLDS_F + ABUF_F) * sizeof(float);

    // iter 0: uniform coupling
    caps_pass<0><<<grid, 256, smem0, stream>>>(x, W, nullptr, nullptr, s0);
    caps_squash<<<(B_TOT * O_TOT + 255) / 256, 256, 0, stream>>>(s0, nullptr, v0);

    // iter 1
    caps_pass<1><<<grid, 256, smem12, stream>>>(x, W, v0, logits, s1);
    caps_squash<<<(B_TOT * O_TOT + 255) / 256, 256, 0, stream>>>(s1, nullptr, v1);

    // iter 2 (final, with bias)
    caps_pass<2><<<grid, 256, smem12, stream>>>(x, W, v1, logits, s2);
    caps_squash<<<(B_TOT * O_TOT + 255) / 256, 256, 0, stream>>>(s2, bias, out);
}